// HCalculator_57183194579314
// MI455X (gfx1250) — compile-verified
//
#include <hip/hip_runtime.h>

#define N_DIM 128

// ---------------------------------------------------------------------------
// Zero-init d_out (harness poisons it with 0xAA before timing).
// b128 stores, fully coalesced.
// ---------------------------------------------------------------------------
__global__ __launch_bounds__(256) void zero_f4_kernel(float4* __restrict__ out,
                                                      int n4) {
    int i = blockIdx.x * blockDim.x + threadIdx.x;
    if (i < n4) {
        out[i] = float4{0.0f, 0.0f, 0.0f, 0.0f};
    }
}

// ---------------------------------------------------------------------------
// One wave32 per edge. Lane l owns feature dims {l, l+32, l+64, l+96}, so
// every global_load_b32 and every global_atomic_add_f32 issued by the wave
// covers a contiguous 128B span -> maximal coalescing of the L2 atomic
// stream. All operands (h, outputs, indices) are L2-resident on MI455X
// (82 MB working set vs 192 MB L2), so atomics resolve in the L2 RMW units.
// ---------------------------------------------------------------------------
__global__ __launch_bounds__(256) void edge_scatter_kernel(
    const float* __restrict__ h,
    const int*   __restrict__ src,
    const int*   __restrict__ dst,
    float*       __restrict__ h_in,    // += h[src] at row dst
    float*       __restrict__ h_out,   // += h[dst] at row src
    int n_edges) {
    const int tid  = blockIdx.x * blockDim.x + threadIdx.x;
    const int e    = tid >> 5;    // wave index == edge index
    const int lane = tid & 31;
    if (e >= n_edges) return;     // wave-uniform guard

    // Prefetch the index stream ahead (gfx1250 global_prefetch_b8;
    // speculative, OOB addresses are silently dropped).
    __builtin_prefetch(src + e + 2048, 0, 1);
    __builtin_prefetch(dst + e + 2048, 0, 1);

    // Indices are wave-uniform: move to SGPRs for scalar-base addressing.
    int s = src[e];
    int d = dst[e];
    s = __builtin_amdgcn_readfirstlane(s);
    d = __builtin_amdgcn_readfirstlane(d);

    const float* hs   = h     + (size_t)s * N_DIM + lane;
    const float* hd   = h     + (size_t)d * N_DIM + lane;
    float*       pin  = h_in  + (size_t)d * N_DIM + lane;
    float*       pout = h_out + (size_t)s * N_DIM + lane;

    float a[4], b[4];
#pragma unroll
    for (int i = 0; i < 4; ++i) {
        a[i] = hs[i * 32];
        b[i] = hd[i * 32];
    }
#pragma unroll
    for (int i = 0; i < 4; ++i) {
        // Non-returning native f32 atomic (global_atomic_add_f32, STOREcnt).
        unsafeAtomicAdd(pin  + i * 32, a[i]);
        unsafeAtomicAdd(pout + i * 32, b[i]);
    }
}

// ---------------------------------------------------------------------------
// Harness entry point.
//   d_in[0] : h   (float32, n_nodes*128)
//   d_in[1] : src (int,     n_edges)
//   d_in[2] : dst (int,     n_edges)
//   d_out   : [h_in (n_nodes*128) | h_out (n_nodes*128)]  float32
// ---------------------------------------------------------------------------
extern "C" void kernel_launch(void* const* d_in, const int* in_sizes, int n_in,
                              void* d_out, int out_size, void* d_ws, size_t ws_size,
                              hipStream_t stream) {
    (void)n_in; (void)d_ws; (void)ws_size;

    const float* h   = (const float*)d_in[0];
    const int*   src = (const int*)d_in[1];
    const int*   dst = (const int*)d_in[2];

    const int n_edges = in_sizes[1];
    const int n_nodes = in_sizes[0] / N_DIM;

    float* h_in  = (float*)d_out;
    float* h_out = (float*)d_out + (size_t)n_nodes * N_DIM;

    // 1) Zero the output accumulators.
    const int n4 = out_size / 4;             // out_size is a multiple of 4
    {
        const int threads = 256;
        const int blocks  = (n4 + threads - 1) / threads;
        zero_f4_kernel<<<blocks, threads, 0, stream>>>((float4*)d_out, n4);
    }

    // 2) Edge scatter-add: one wave32 per edge, 8 edges per 256-thread block.
    {
        const int threads      = 256;
        const long long nthr   = (long long)n_edges * 32;
        const int blocks       = (int)((nthr + threads - 1) / threads);
        edge_scatter_kernel<<<blocks, threads, 0, stream>>>(
            h, src, dst, h_in, h_out, n_edges);
    }
}